// SparseGroupedConv2d_8151847927920
// MI455X (gfx1250) — compile-verified
//
#include <hip/hip_runtime.h>

typedef float v2f __attribute__((ext_vector_type(2)));
typedef float v8f __attribute__((ext_vector_type(8)));

#define BATCH   16
#define CIN     256
#define COUT    256
#define NGROUPS 2
#define CPG     128          // channels per group (in == out here)
#define IMH     112
#define IMW     112
#define HW      (IMH * IMW)  // 12544
#define NTOT    (BATCH * HW) // 200704
#define KHW     9            // 3x3

#define M_TILE   64
#define N_TILE   128
#define KC       16          // channels staged per chunk
#define NCHUNK   72          // 9 taps * 8 channel chunks
#define A_STRIDE 20          // even (b64-aligned pairs); 20m mod 64 distinct, no cross-half alias
#define B_STRIDE 288         // 128 pairs*2 + 32 pad; 288 mod 64 == 32 -> halves in disjoint banks

__global__ __launch_bounds__(256)
void grouped_conv2d_wmma_f32(const float* __restrict__ x,
                             const float* __restrict__ w,
                             const float* __restrict__ bias,
                             float* __restrict__ y)
{
    __shared__ __align__(16) float Alds[2][M_TILE * A_STRIDE]; // 64 x 16 weight slice
    __shared__ __align__(16) float Blds[2][8 * B_STRIDE];      // 8 K-pairs x 128 positions x 2

    const int tid  = threadIdx.x;
    const int lane = tid & 31;
    const int wave = tid >> 5;
    const int mw   = wave >> 1;   // 0..3  (M sub-tile)
    const int nw   = wave & 1;    // 0..1  (N sub-tile)
    const int lr   = lane & 15;
    const int hi   = lane >> 4;

    const int nTile = blockIdx.x; // 0..1567
    const int mTile = blockIdx.y; // 0..1
    const int g     = blockIdx.z; // 0..1
    const int nBase = nTile * N_TILE;

    // --- B (activation) staging map: one spatial position per thread, 8 channel rows
    const int nLoc  = tid & 127;
    const int cHalf = tid >> 7;
    const int nGlob = nBase + nLoc;
    const int bIdx  = nGlob / HW;
    const int rem   = nGlob % HW;
    const int oh    = rem / IMW;
    const int ow    = rem % IMW;
    const float* xb = x + ((size_t)bIdx * CIN + (size_t)g * CPG) * HW;

    // --- A (weight) staging map: one output channel per 4 threads, 4 channels each
    const int amRow = tid >> 2;                 // 0..63
    const int ac4   = (tid & 3) * 4;            // 0,4,8,12
    const float* wRow = w + (size_t)(g * CPG + mTile * M_TILE + amRow) * (CPG * KHW);

    v8f zero = {0.f, 0.f, 0.f, 0.f, 0.f, 0.f, 0.f, 0.f};
    v8f acc[4];
#pragma unroll
    for (int f = 0; f < 4; ++f) acc[f] = zero;

    float aReg[4];
    float bReg[8];

    auto loadChunk = [&](int t) {
        const int tap = t >> 3;
        const int c0  = (t & 7) << 4;
        const int kh  = tap / 3;
        const int kw  = tap - kh * 3;
        const int ih  = oh + kh - 1;
        const int iw  = ow + kw - 1;
        const bool valid = ((unsigned)ih < (unsigned)IMH) && ((unsigned)iw < (unsigned)IMW);
#pragma unroll
        for (int j = 0; j < 4; ++j)
            aReg[j] = wRow[(size_t)(c0 + ac4 + j) * KHW + kh * 3 + kw];
#pragma unroll
        for (int i = 0; i < 8; ++i) {
            const int cLoc = cHalf * 8 + i;
            bReg[i] = valid ? xb[(size_t)(c0 + cLoc) * HW + ih * IMW + iw] : 0.0f;
        }
    };

    auto storeChunk = [&](int buf) {
#pragma unroll
        for (int j = 0; j < 4; ++j)
            Alds[buf][amRow * A_STRIDE + ac4 + j] = aReg[j];
#pragma unroll
        for (int i = 0; i < 8; ++i) {
            const int cLoc = cHalf * 8 + i;
            // K-pair interleaved, fragment-ready: one ds_load_b64 per B frag
            Blds[buf][(cLoc >> 1) * B_STRIDE + nLoc * 2 + (cLoc & 1)] = bReg[i];
        }
    };

    loadChunk(0);
    storeChunk(0);
    __syncthreads();

    for (int t = 0; t < NCHUNK; ++t) {
        const int cur  = t & 1;
        const bool more = (t + 1) < NCHUNK;
        if (more) loadChunk(t + 1);   // global loads in flight during WMMAs

#pragma unroll
        for (int ks = 0; ks < 4; ++ks) {
            v2f a = *(const v2f*)&Alds[cur][(mw * 16 + lr) * A_STRIDE + ks * 4 + hi * 2];
#pragma unroll
            for (int f = 0; f < 4; ++f) {
                v2f b = *(const v2f*)&Blds[cur][(ks * 2 + hi) * B_STRIDE
                                               + (nw * 64 + f * 16 + lr) * 2];
                acc[f] = __builtin_amdgcn_wmma_f32_16x16x4_f32(
                    false, a, false, b, (short)0, acc[f], false, false);
            }
        }

        __syncthreads();                    // all waves done reading Alds/Blds[cur]
        if (more) storeChunk(1 - cur);      // commit next chunk to alternate buffer
        __syncthreads();                    // stores visible before next compute
    }

    // store: C layout -> VGPR r holds M=r (lanes 0-15) / M=r+8 (lanes 16-31), N=lr
#pragma unroll
    for (int f = 0; f < 4; ++f) {
        const int nOut = nBase + nw * 64 + f * 16 + lr;
        const int b2   = nOut / HW;
        const int r2   = nOut % HW;
        const int oh2  = r2 / IMW;
        const int ow2  = r2 % IMW;
#pragma unroll
        for (int r = 0; r < 8; ++r) {
            const int mLoc = mw * 16 + hi * 8 + r;
            const int oc   = g * CPG + mTile * M_TILE + mLoc;
            const size_t idx = ((size_t)b2 * COUT + oc) * HW + oh2 * IMW + ow2;
            y[idx] = acc[f][r] + bias[oc];
        }
    }
}

extern "C" void kernel_launch(void* const* d_in, const int* in_sizes, int n_in,
                              void* d_out, int out_size, void* d_ws, size_t ws_size,
                              hipStream_t stream) {
    (void)in_sizes; (void)n_in; (void)d_ws; (void)ws_size; (void)out_size;
    const float* x    = (const float*)d_in[0];
    const float* w    = (const float*)d_in[1];
    const float* bias = (const float*)d_in[2];
    float* y          = (float*)d_out;

    dim3 grid(NTOT / N_TILE, CPG / M_TILE, NGROUPS); // (1568, 2, 2)
    dim3 block(256);
    grouped_conv2d_wmma_f32<<<grid, block, 0, stream>>>(x, w, bias, y);
}